// CachedHeavyRecentAttentionMasker_23536420782086
// MI455X (gfx1250) — compile-verified
//
#include <hip/hip_runtime.h>
#include <stdint.h>

// Problem constants (fixed by the reference: (2,16,2048,2048) fp32)
#define BS        2
#define NHEAD     16
#define BH        (BS * NHEAD)        // 32
#define QLEN      2048
#define KLEN      2048
#define HEAVY     204                 // int(0.1 * 2048)
#define RECENT    204
#define MINVAL    (-3.402823466e38f)  // jnp.finfo(f32).min == -FLT_MAX

#define QCHUNKS   32                  // blocks per (b,h) in phase 1
#define ROWS      (QLEN / QCHUNKS)    // 64 rows per block
#define P1_THREADS 256                // 8 waves (wave32)

typedef float v4f __attribute__((ext_vector_type(4)));

// ---------------------------------------------------------------------------
// Issue async copy of one 8 KB row (2048 f32) into LDS.
// Thread t moves bytes [16t,16t+16) and [16t+4096, ...) -> 2x b128 per thread.
// CDNA5 async-tensor path: GLOBAL_LOAD_ASYNC_TO_LDS_B128 (tracked by ASYNCcnt).
// ---------------------------------------------------------------------------
__device__ __forceinline__ void issue_row_async(const float* rowptr,
                                                const float* lds_dst, int t) {
  // Generic LDS pointer: low 32 bits are the wave-relative LDS byte address
  // (addrspacecast generic->local == truncate on AMDGPU).
  unsigned lds0 = (unsigned)(uintptr_t)lds_dst + (unsigned)(t * 16);
  unsigned g0   = (unsigned)(t * 16);
  asm volatile("global_load_async_to_lds_b128 %0, %1, %2"
               :: "v"(lds0), "v"(g0), "s"(rowptr) : "memory");
  unsigned lds1 = lds0 + 4096u;
  unsigned g1   = g0 + 4096u;
  asm volatile("global_load_async_to_lds_b128 %0, %1, %2"
               :: "v"(lds1), "v"(g1), "s"(rowptr) : "memory");
}

__device__ __forceinline__ float blk_reduce_max(float v, volatile float* red, int t) {
#pragma unroll
  for (int off = 16; off > 0; off >>= 1)
    v = fmaxf(v, __shfl_xor(v, off, 32));
  __syncthreads();                       // red[] safe to overwrite
  if ((t & 31) == 0) red[t >> 5] = v;
  __syncthreads();
  float r = red[0];
#pragma unroll
  for (int i = 1; i < P1_THREADS / 32; ++i) r = fmaxf(r, red[i]);
  return r;
}

__device__ __forceinline__ float blk_reduce_sum(float v, volatile float* red, int t) {
#pragma unroll
  for (int off = 16; off > 0; off >>= 1)
    v += __shfl_xor(v, off, 32);
  __syncthreads();
  if ((t & 31) == 0) red[t >> 5] = v;
  __syncthreads();
  float r = red[0];
#pragma unroll
  for (int i = 1; i < P1_THREADS / 32; ++i) r += red[i];
  return r;
}

// ---------------------------------------------------------------------------
// Phase 1: softmax column sums. colsum[bh][k] += softmax(row q)[k] over q.
// Async double-buffered row staging through LDS.
// ---------------------------------------------------------------------------
__global__ void __launch_bounds__(P1_THREADS)
hr_phase1_colsum(const float* __restrict__ in, float* __restrict__ colsum) {
  __shared__ float rowbuf[2][KLEN];          // 16 KB double buffer
  __shared__ float red[P1_THREADS / 32];

  const int t     = threadIdx.x;
  const int bh    = blockIdx.x / QCHUNKS;
  const int chunk = blockIdx.x % QCHUNKS;
  const int q0    = chunk * ROWS;

  const float* base = in + (size_t)(bh * QLEN + q0) * KLEN;

  float acc[8];
#pragma unroll
  for (int j = 0; j < 8; ++j) acc[j] = 0.0f;

  // Prefetch row 0
  issue_row_async(base, &rowbuf[0][0], t);

  for (int r = 0; r < ROWS; ++r) {
    if (r + 1 < ROWS)
      issue_row_async(base + (size_t)(r + 1) * KLEN, &rowbuf[(r + 1) & 1][0], t);

    // Wait until row r's two b128s landed; row r+1's (if any) stay in flight.
    if (r + 1 < ROWS)
      asm volatile("s_wait_asynccnt 0x2" ::: "memory");
    else
      asm volatile("s_wait_asynccnt 0x0" ::: "memory");
    __syncthreads();                       // row r visible to all waves

    const float* row = &rowbuf[r & 1][0];
    float v[8];
#pragma unroll
    for (int j = 0; j < 8; ++j) v[j] = row[t + (j << 8)];   // bank-conflict-free

    float m = v[0];
#pragma unroll
    for (int j = 1; j < 8; ++j) m = fmaxf(m, v[j]);
    const float rowmax = blk_reduce_max(m, red, t);

    float e[8], s = 0.0f;
#pragma unroll
    for (int j = 0; j < 8; ++j) { e[j] = __expf(v[j] - rowmax); s += e[j]; }
    const float rowsum = blk_reduce_sum(s, red, t);
    const float inv = 1.0f / rowsum;

#pragma unroll
    for (int j = 0; j < 8; ++j) acc[j] += e[j] * inv;

    __syncthreads();                       // all reads of rowbuf[r&1] done
  }

#pragma unroll
  for (int j = 0; j < 8; ++j)
    atomicAdd(&colsum[bh * KLEN + t + (j << 8)], acc[j]);
}

// ---------------------------------------------------------------------------
// Phase 2: per (b,h) iterative top-204 (tie-break: smallest index, matching
// jax.lax.top_k). One wave32 per (b,h); emits 2048-bit keep mask.
// ---------------------------------------------------------------------------
__global__ void __launch_bounds__(32)
hr_phase2_topk(const float* __restrict__ colsum, unsigned* __restrict__ maskbits) {
  __shared__ float    vals[KLEN];
  __shared__ float    candv[32];
  __shared__ int      candi[32];
  __shared__ unsigned bits[KLEN / 32];

  const int t  = threadIdx.x;
  const int bh = blockIdx.x;

  for (int j = t; j < KLEN; j += 32) vals[j] = colsum[bh * KLEN + j];
  for (int j = t; j < KLEN / 32; j += 32) bits[j] = 0u;
  __syncthreads();

  for (int it = 0; it < HEAVY; ++it) {
    float bv = MINVAL; int bi = 0x7fffffff;
#pragma unroll 4
    for (int j = 0; j < KLEN / 32; ++j) {
      int idx = t + (j << 5);              // increasing -> lane-local min-index tie-break
      float v = vals[idx];
      if (v > bv) { bv = v; bi = idx; }
    }
    candv[t] = bv; candi[t] = bi;
    __syncthreads();
    if (t == 0) {
      float gv = candv[0]; int gi = candi[0];
      for (int l = 1; l < 32; ++l) {
        if (candv[l] > gv || (candv[l] == gv && candi[l] < gi)) {
          gv = candv[l]; gi = candi[l];
        }
      }
      vals[gi] = MINVAL;                   // remove from future rounds
      bits[gi >> 5] |= 1u << (gi & 31);
    }
    __syncthreads();
  }

  for (int j = t; j < KLEN / 32; j += 32) maskbits[(bh << 6) + j] = bits[j];
}

// ---------------------------------------------------------------------------
// Phase 3: streaming mask apply. Nontemporal B128 both ways (1 GB >> 192 MB L2).
// ---------------------------------------------------------------------------
__global__ void __launch_bounds__(256)
hr_phase3_apply(const float* __restrict__ in, const unsigned* __restrict__ maskbits,
                float* __restrict__ out) {
  const size_t gid = (size_t)blockIdx.x * blockDim.x + threadIdx.x; // float4 index
  const int    k4    = (int)((gid & (KLEN / 4 - 1)) << 2);
  const size_t rowid = gid >> 9;                                    // / (KLEN/4)
  const int    q     = (int)(rowid & (QLEN - 1));
  const int    bh    = (int)(rowid >> 11);

  v4f v = __builtin_nontemporal_load((const v4f*)in + gid);
  const unsigned w = maskbits[(bh << 6) + (k4 >> 5)];               // 8 KB total, hot in cache

  v4f r;
#pragma unroll
  for (int j = 0; j < 4; ++j) {
    const int k  = k4 + j;
    const int d  = k - q;
    const bool keep = (((w >> (k & 31)) & 1u) != 0u) | (d <= RECENT && d >= -RECENT);
    r[j] = keep ? v[j] : MINVAL;
  }
  __builtin_nontemporal_store(r, (v4f*)out + gid);
}

// ---------------------------------------------------------------------------
extern "C" void kernel_launch(void* const* d_in, const int* in_sizes, int n_in,
                              void* d_out, int out_size, void* d_ws, size_t ws_size,
                              hipStream_t stream) {
  const float* in  = (const float*)d_in[0];
  float*       out = (float*)d_out;

  float*    colsum   = (float*)d_ws;                                  // 32*2048 f32 = 256 KB
  unsigned* maskbits = (unsigned*)((char*)d_ws + (size_t)BH * KLEN * sizeof(float)); // 8 KB

  hipMemsetAsync(colsum, 0, (size_t)BH * KLEN * sizeof(float), stream);

  hr_phase1_colsum<<<BH * QCHUNKS, P1_THREADS, 0, stream>>>(in, colsum);
  hr_phase2_topk<<<BH, 32, 0, stream>>>(colsum, maskbits);

  const size_t total4 = (size_t)BH * QLEN * KLEN / 4;                 // 33,554,432
  hr_phase3_apply<<<(unsigned)(total4 / 256), 256, 0, stream>>>(in, maskbits, out);
}